// ATACSplitPool_41824391528702
// MI455X (gfx1250) — compile-verified
//
#include <hip/hip_runtime.h>
#include <hip/hip_bf16.h>
#include <math.h>

// Problem constants (fixed by setup_inputs)
#define B_      4
#define L_      100000
#define D_      639
#define NP_     400
#define PATCH_  25
#define LP_     4000        // L / PATCH
#define CIN_    655         // D + 16
#define KTOT_   1965        // CIN * 3
#define KPAD_   1968        // next multiple of 4
#define BN_EPS_ 1e-5f

typedef __attribute__((ext_vector_type(2))) float v2f;
typedef __attribute__((ext_vector_type(8))) float v8f;

// ---------------------------------------------------------------------------
// 0) Per-sample prefix scan of peak_split (in patch units).
//    seg_pb[b][s] = first patch of segment s; stride NP_+2 per sample.
// ---------------------------------------------------------------------------
__global__ void seg_prefix(const int* __restrict__ split, int* __restrict__ segpb) {
  int b = threadIdx.x;
  if (b < B_) {
    int acc = 0;
    segpb[b * (NP_ + 2) + 0] = 0;
    for (int s = 0; s < NP_ + 1; ++s) {
      acc += split[b * (NP_ + 1) + s] / PATCH_;
      segpb[b * (NP_ + 2) + s + 1] = acc;
    }
  }
}

// ---------------------------------------------------------------------------
// 1) THE bandwidth kernel: one streaming pass over x (1.02 GB).
//    Block = one (b, patch). Produces per-patch sum (for segment means) and
//    per-patch max (MaxPool1d(25,25) for the joint path) in one read.
//    Lanes cover consecutive d -> fully coalesced b32 loads.
// ---------------------------------------------------------------------------
__global__ void __launch_bounds__(256)
patch_reduce(const float* __restrict__ x, float* __restrict__ psum,
             float* __restrict__ pmax) {
  int blk = blockIdx.x;                               // b*LP_ + p
  const float* base = x + (size_t)blk * PATCH_ * D_;  // (b*L + p*25)*D
  for (int d = threadIdx.x; d < D_; d += 256) {
    float s = 0.0f, m = -INFINITY;
#pragma unroll 5
    for (int r = 0; r < PATCH_; ++r) {
      float v = base[(size_t)r * D_ + d];
      s += v;
      m = fmaxf(m, v);
    }
    size_t o = (size_t)blk * D_ + d;
    psum[o] = s;
    pmax[o] = m;
  }
}

// ---------------------------------------------------------------------------
// 2) atac: max-pool(25) then log10(1+.)  (log10 is monotonic, so
//    max(log10(a+1)) == log10(max(a)+1) — bitwise identical, 1 log per patch)
// ---------------------------------------------------------------------------
__global__ void atac_pool(const float* __restrict__ atac, float* __restrict__ apool) {
  int idx = blockIdx.x * blockDim.x + threadIdx.x;    // b*LP_ + p
  if (idx >= B_ * LP_) return;
  int b = idx / LP_, p = idx - b * LP_;
  const float* src = atac + (size_t)b * L_ + (size_t)p * PATCH_;
  float m = src[0];
#pragma unroll
  for (int r = 1; r < PATCH_; ++r) m = fmaxf(m, src[r]);
  apool[idx] = log10f(m + 1.0f);
}

// ---------------------------------------------------------------------------
// 3) Transpose joint_w (16, 1965) -> wT (KPAD_, 16), zero-padded K tail so the
//    WMMA K loop needs no remainder handling (pad weights = 0 => no effect).
// ---------------------------------------------------------------------------
__global__ void wt_build(const float* __restrict__ jw, float* __restrict__ wT) {
  int idx = blockIdx.x * blockDim.x + threadIdx.x;    // k*16 + m
  if (idx >= KPAD_ * 16) return;
  int k = idx >> 4, m = idx & 15;
  wT[idx] = (k < KTOT_) ? jw[m * KTOT_ + k] : 0.0f;
}

// ---------------------------------------------------------------------------
// 4) atac conv(16,1,3) batch-norm stats (training mode, biased var) —
//    deterministic LDS tree reduction, one block per channel.
// ---------------------------------------------------------------------------
__global__ void __launch_bounds__(256)
atac_stats(const float* __restrict__ apool, const float* __restrict__ aw,
           float* __restrict__ stats) {
  __shared__ float ssum[256], ssq[256];
  int ch = blockIdx.x, tid = threadIdx.x;
  float w0 = aw[ch * 3 + 0], w1 = aw[ch * 3 + 1], w2 = aw[ch * 3 + 2];
  float s = 0.0f, q = 0.0f;
  for (int t = tid; t < B_ * LP_; t += 256) {
    int b = t / LP_, lp = t - b * LP_;
    const float* row = apool + b * LP_;
    float v = w1 * row[lp];
    if (lp > 0)        v += w0 * row[lp - 1];
    if (lp < LP_ - 1)  v += w2 * row[lp + 1];
    s += v; q += v * v;
  }
  ssum[tid] = s; ssq[tid] = q;
  __syncthreads();
  for (int off = 128; off > 0; off >>= 1) {
    if (tid < off) { ssum[tid] += ssum[tid + off]; ssq[tid] += ssq[tid + off]; }
    __syncthreads();
  }
  if (tid == 0) {
    float mean = ssum[0] / (float)(B_ * LP_);
    float var  = ssq[0] / (float)(B_ * LP_) - mean * mean;
    stats[ch] = mean;
    stats[16 + ch] = rsqrtf(var + BN_EPS_);
  }
}

// ---------------------------------------------------------------------------
// 5) Recompute atac conv, apply BN+ReLU, store ap as (B, Lp, 16).
// ---------------------------------------------------------------------------
__global__ void atac_bn_relu(const float* __restrict__ apool, const float* __restrict__ aw,
                             const float* __restrict__ stats, float* __restrict__ appost) {
  int idx = blockIdx.x * blockDim.x + threadIdx.x;    // (b*LP+lp)*16 + ch
  if (idx >= B_ * LP_ * 16) return;
  int ch = idx & 15, t = idx >> 4;
  int b = t / LP_, lp = t - b * LP_;
  const float* row = apool + b * LP_;
  float v = aw[ch * 3 + 1] * row[lp];
  if (lp > 0)        v += aw[ch * 3 + 0] * row[lp - 1];
  if (lp < LP_ - 1)  v += aw[ch * 3 + 2] * row[lp + 1];
  v = (v - stats[ch]) * stats[16 + ch];
  appost[idx] = v > 0.0f ? v : 0.0f;
}

// ---------------------------------------------------------------------------
// 6) Joint conv as WMMA GEMM: out(16 ch, 16 lp) = W(16, K) x im2col(K, 16 lp).
//    One wave per 16-lp tile. im2col tile (656 rows x 18 lp = 47 KB) staged in
//    LDS (conflict-free: row stride 18 floats -> 32 distinct banks).
//    K consumed 4 at a time with V_WMMA_F32_16X16X4_F32 (fp32 == reference).
//    A frag (16x4): lanes 0-15 hold K=k0,k0+1; lanes 16-31 hold K=k0+2,k0+3.
//    B frag (4x16): mirrored K split, N = lane & 15.
// ---------------------------------------------------------------------------
__global__ void __launch_bounds__(32)
joint_conv_wmma(const float* __restrict__ pmax, const float* __restrict__ appost,
                const float* __restrict__ wT, float* __restrict__ jpre) {
  __shared__ float tile[CIN_ + 1][18];                // 47232 B
  const int blk  = blockIdx.x;
  const int b    = blk / (LP_ / 16);
  const int lp0  = (blk % (LP_ / 16)) * 16;
  const int lane = threadIdx.x;

  // Stage im2col tile (with SAME halo, zero outside [0, Lp)) into LDS.
  for (int j = 0; j < 18; ++j) {
    int  lpg   = lp0 + j - 1;
    bool valid = (lpg >= 0) && (lpg < LP_);
    int  lpc   = valid ? lpg : 0;
    const float* xrow = pmax   + ((size_t)(b * LP_ + lpc)) * D_;
    const float* arow = appost + ((size_t)(b * LP_ + lpc)) * 16;
    for (int c = lane; c < D_; c += 32) tile[c][j] = valid ? xrow[c] : 0.0f;
    if (lane < 16) tile[D_ + lane][j] = valid ? arow[lane] : 0.0f;
    if (lane == 0) tile[CIN_][j] = 0.0f;              // zero row for K padding
  }
  __syncthreads();

  const int half = lane >> 4;                         // 0: K=k0..k0+1, 1: K=k0+2..k0+3
  const int m    = lane & 15;                         // A: out-channel / B: lp column
  v8f acc = {};
  for (int k0 = 0; k0 < KPAD_; k0 += 4) {
    int ka = k0 + half * 2;
    v2f a, bb;
    a.x = wT[ka * 16 + m];
    a.y = wT[(ka + 1) * 16 + m];
    int c0 = ka / 3,       r0 = ka - c0 * 3;          // K -> (cin, conv tap)
    int c1 = (ka + 1) / 3, r1 = (ka + 1) - c1 * 3;
    bb.x = tile[c0][m + r0];
    bb.y = tile[c1][m + r1];
    acc = __builtin_amdgcn_wmma_f32_16x16x4_f32(false, a, false, bb,
                                                (short)0, acc, false, false);
  }

  // C/D layout: VGPR r -> M = r (lanes 0-15) or r+8 (lanes 16-31); N = lane&15.
  const int lp = lp0 + m;
  float* outp = jpre + ((size_t)(b * LP_ + lp)) * 16 + half * 8;
#pragma unroll
  for (int r = 0; r < 8; ++r) outp[r] = acc[r];
}

// ---------------------------------------------------------------------------
// 7) Joint BN stats (same deterministic reduction, reads pre-BN conv output).
// ---------------------------------------------------------------------------
__global__ void __launch_bounds__(256)
joint_stats(const float* __restrict__ jpre, float* __restrict__ stats) {
  __shared__ float ssum[256], ssq[256];
  int ch = blockIdx.x, tid = threadIdx.x;
  float s = 0.0f, q = 0.0f;
  for (int t = tid; t < B_ * LP_; t += 256) {
    float v = jpre[(size_t)t * 16 + ch];
    s += v; q += v * v;
  }
  ssum[tid] = s; ssq[tid] = q;
  __syncthreads();
  for (int off = 128; off > 0; off >>= 1) {
    if (tid < off) { ssum[tid] += ssum[tid + off]; ssq[tid] += ssq[tid + off]; }
    __syncthreads();
  }
  if (tid == 0) {
    float mean = ssum[0] / (float)(B_ * LP_);
    float var  = ssq[0] / (float)(B_ * LP_) - mean * mean;
    stats[ch] = mean;
    stats[16 + ch] = rsqrtf(var + BN_EPS_);
  }
}

// ---------------------------------------------------------------------------
// 8) x_region: per-(b,seg) mean of x from patch sums -> out[..., 0:639].
// ---------------------------------------------------------------------------
__global__ void __launch_bounds__(256)
x_region(const float* __restrict__ psum, const int* __restrict__ segpb,
         float* __restrict__ out) {
  int blk = blockIdx.x;                               // b*NP_ + s
  int b = blk / NP_, s = blk - b * NP_;
  int ps = segpb[b * (NP_ + 2) + s];
  int pe = segpb[b * (NP_ + 2) + s + 1];
  float inv = 1.0f / (float)((pe - ps) * PATCH_);
  for (int d = threadIdx.x; d < D_; d += 256) {
    float acc = 0.0f;
    for (int p = ps; p < pe; ++p) acc += psum[((size_t)(b * LP_ + p)) * D_ + d];
    out[((size_t)(b * NP_ + s)) * CIN_ + d] = acc * inv;
  }
}

// ---------------------------------------------------------------------------
// 9) joint_region: BN+ReLU, per-(b,seg) mean over patches, log2(1+.)
//    -> out[..., 639:655].
// ---------------------------------------------------------------------------
__global__ void __launch_bounds__(32)
joint_region(const float* __restrict__ jpre, const float* __restrict__ stats,
             const int* __restrict__ segpb, float* __restrict__ out) {
  int blk = blockIdx.x;                               // b*NP_ + s
  int b = blk / NP_, s = blk - b * NP_;
  int lane = threadIdx.x;
  if (lane >= 16) return;
  int ps = segpb[b * (NP_ + 2) + s];
  int pe = segpb[b * (NP_ + 2) + s + 1];
  float mean = stats[lane], istd = stats[16 + lane];
  float acc = 0.0f;
  for (int p = ps; p < pe; ++p) {
    float v = (jpre[((size_t)(b * LP_ + p)) * 16 + lane] - mean) * istd;
    acc += v > 0.0f ? v : 0.0f;
  }
  float mv = acc / (float)(pe - ps);
  out[((size_t)(b * NP_ + s)) * CIN_ + D_ + lane] = log2f(mv + 1.0f);
}

// ---------------------------------------------------------------------------
extern "C" void kernel_launch(void* const* d_in, const int* in_sizes, int n_in,
                              void* d_out, int out_size, void* d_ws, size_t ws_size,
                              hipStream_t stream) {
  (void)in_sizes; (void)n_in; (void)out_size; (void)ws_size;
  const float* x          = (const float*)d_in[0];
  const float* atac       = (const float*)d_in[1];
  const float* atac_w     = (const float*)d_in[2];
  const float* joint_w    = (const float*)d_in[3];
  const int*   peak_split = (const int*)d_in[4];
  float* out = (float*)d_out;

  // Workspace carve-up (~84 MB, every region fully written before read).
  float* ws      = (float*)d_ws;
  float* pm_sum  = ws;                                  // B*LP*D
  float* pm_max  = pm_sum  + (size_t)B_ * LP_ * D_;     // B*LP*D
  float* a_pool  = pm_max  + (size_t)B_ * LP_ * D_;     // B*LP
  float* ap_post = a_pool  + (size_t)B_ * LP_;          // B*LP*16
  float* wT      = ap_post + (size_t)B_ * LP_ * 16;     // KPAD*16
  float* j_pre   = wT      + (size_t)KPAD_ * 16;        // B*LP*16
  float* stats_a = j_pre   + (size_t)B_ * LP_ * 16;     // 32
  float* stats_j = stats_a + 32;                        // 32
  int*   seg_pb  = (int*)(stats_j + 32);                // B*(NP_+2)

  seg_prefix     <<<1, 32, 0, stream>>>(peak_split, seg_pb);
  patch_reduce   <<<B_ * LP_, 256, 0, stream>>>(x, pm_sum, pm_max);
  atac_pool      <<<(B_ * LP_ + 255) / 256, 256, 0, stream>>>(atac, a_pool);
  wt_build       <<<(KPAD_ * 16 + 255) / 256, 256, 0, stream>>>(joint_w, wT);
  atac_stats     <<<16, 256, 0, stream>>>(a_pool, atac_w, stats_a);
  atac_bn_relu   <<<(B_ * LP_ * 16 + 255) / 256, 256, 0, stream>>>(a_pool, atac_w, stats_a, ap_post);
  joint_conv_wmma<<<B_ * (LP_ / 16), 32, 0, stream>>>(pm_max, ap_post, wT, j_pre);
  joint_stats    <<<16, 256, 0, stream>>>(j_pre, stats_j);
  x_region       <<<B_ * NP_, 256, 0, stream>>>(pm_sum, seg_pb, out);
  joint_region   <<<B_ * NP_, 32, 0, stream>>>(j_pre, stats_j, seg_pb, out);
}